// LSTM_67542655696957
// MI455X (gfx1250) — compile-verified
//
#include <hip/hip_runtime.h>

typedef __attribute__((ext_vector_type(2))) float v2f;
typedef __attribute__((ext_vector_type(8))) float v8f;
typedef unsigned int u32;
typedef __attribute__((ext_vector_type(4))) u32 v4u;
typedef __attribute__((ext_vector_type(4))) int  v4i;
typedef __attribute__((ext_vector_type(8))) int  v8i;

#define BATCH   64
#define SEQ_T   2048
#define FEAT    256
#define HID     256
#define G4      1024
#define NBLOCKS 64          // 4 mt-groups x 16 ht-blocks
#define TPB     128         // 4 waves: one gate-block (i,f,g,o) each
#define GROUP_BLOCKS 16

// LDS layout (dynamic): x tile, h tile (16 rows x 260 padded floats), gate tiles
#define LDS_ROW   260                        // 256 + 4 pad dwords -> conflict-free ds_load_b64
#define LDSX_OFF  0                          // bytes
#define LDSH_OFF  (16 * LDS_ROW * 4)         // 16640
#define LDSG_OFF  (2 * 16 * LDS_ROW * 4)     // 33280
#define GPITCH    (16 * 17)                  // per-gate-tile floats (row pad 17)
#define LDS_BYTES (LDSG_OFF + 4 * GPITCH * 4)

// ws layout (floats): 4 per-group barrier counters (256B apart), then h (64x256)
#define WS_CTR_STRIDE 64
#define WS_H_OFF      256
#define WS_ZERO_BYTES ((WS_H_OFF + BATCH * HID) * sizeof(float))

#if __has_builtin(__builtin_amdgcn_tensor_load_to_lds) && __has_builtin(__builtin_amdgcn_s_wait_tensorcnt)
#define USE_TDM 1
#else
#define USE_TDM 0
#endif

__device__ __forceinline__ float fast_sigmoid(float z) {
    return 1.0f / (1.0f + __expf(-z));
}
__device__ __forceinline__ float fast_tanh(float z) {
    return 2.0f / (1.0f + __expf(-2.0f * z)) - 1.0f;
}

// ---- Tensor Data Mover descriptors (CDNA5 D#, CDNA5 ISA) ----
// 2D tile: 16 rows x 256 fp32, LDS pad 4 dwords per 256 (pad_interval=7, pad_amount=3)
__device__ __forceinline__ v4u tdm_g0(u32 lds_byte_off, const void* gptr) {
    unsigned long long ga = (unsigned long long)(uintptr_t)gptr;
    v4u g;
    g.x = 1u;                                  // count=1, user mode
    g.y = lds_byte_off;                        // lds_addr [63:32]
    g.z = (u32)ga;                             // global_addr [95:64]
    g.w = (u32)(ga >> 32) | 0x80000000u;       // global_addr hi + type=2 [127:126]
    return g;
}
__device__ __forceinline__ v8i tdm_g1(u32 row_stride_elems) {
    v8i g;
    g[0] = (int)((2u << 16) | (1u << 20) | (7u << 22) | (3u << 25)); // data_size=4B, pad_en, int=256dw, amt=4dw
    g[1] = (int)(256u << 16);     // tensor_dim0 = 256   (bits 79:48, low half)
    g[2] = (int)(16u  << 16);     // tensor_dim1 = 16    (bits 111:80, low half)
    g[3] = (int)(256u << 16);     // tile_dim0   = 256   (bits 127:112)
    g[4] = 16;                    // tile_dim1   = 16    (bits 143:128)
    g[5] = (int)row_stride_elems; // tensor_dim0_stride low32 (bits 207:160)
    g[6] = 0;
    g[7] = 0;
    return g;
}
__device__ __forceinline__ void tdm_issue(v4u g0, v8i g1) {
#if USE_TDM
    v4i z4 = {0, 0, 0, 0};
#if defined(__clang_major__) && (__clang_major__ >= 23)
    v8i z8 = {0, 0, 0, 0, 0, 0, 0, 0};
    __builtin_amdgcn_tensor_load_to_lds(g0, g1, z4, z4, z8, 0);
#else
    __builtin_amdgcn_tensor_load_to_lds(g0, g1, z4, z4, 0);
#endif
#endif
}

#if !USE_TDM
// Fallback: cooperative vector copy of a 16 x 256 fp32 tile into padded LDS
__device__ __forceinline__ void coop_copy_tile(float* dst, const float* src, size_t row_stride) {
    for (int i = threadIdx.x; i < 16 * 64; i += TPB) {
        const int r = i >> 6, c = (i & 63) * 4;
        const float4 v = *(const float4*)(src + (size_t)r * row_stride + c);
        *(float4*)(dst + r * LDS_ROW + c) = v;
    }
}
#endif

// 16-block group barrier (monotonic counter, memset to 0 on stream before launch)
__device__ __forceinline__ void group_barrier(unsigned* ctr, unsigned target) {
    __syncthreads();
    if (threadIdx.x == 0) {
        __hip_atomic_fetch_add(ctr, 1u, __ATOMIC_RELEASE, __HIP_MEMORY_SCOPE_AGENT);
        while (__hip_atomic_load(ctr, __ATOMIC_ACQUIRE, __HIP_MEMORY_SCOPE_AGENT) < target)
            __builtin_amdgcn_s_sleep(1);
    }
    __syncthreads();
}

__global__ __launch_bounds__(TPB, 1)
void lstm_persistent(const float* __restrict__ x,     // (64, 2048, 256)
                     const float* __restrict__ Wx,    // (256, 1024)
                     const float* __restrict__ Wh,    // (256, 1024)
                     const float* __restrict__ Bx,    // (1024)
                     const float* __restrict__ Bh,    // (1024)
                     float*       __restrict__ out,   // (64,2048,256) then h_n (64,256)
                     float*       __restrict__ ws)
{
    extern __shared__ float smem[];
    float* sh_x = smem;                      // 16 x 260
    float* sh_h = smem + 16 * LDS_ROW;       // 16 x 260
    float* sh_g = smem + 2 * 16 * LDS_ROW;   // 4 x 16 x 17

    const int mt = blockIdx.x >> 4;          // batch tile 0..3
    const int ht = blockIdx.x & 15;          // hidden tile 0..15
    const int w  = threadIdx.x >> 5;         // gate block 0..3 (i,f,g,o)
    const int lane = threadIdx.x & 31;
    const int l16  = lane & 15;
    const int half = lane >> 4;
    const int koff = half * 2;               // fp32 WMMA A/B K sub-offset
    const int m0 = mt * 16;
    const int n0 = w * HID + ht * 16;        // column in 4H gate space

    unsigned* ctr = (unsigned*)ws + (size_t)mt * WS_CTR_STRIDE;
    float* hbuf   = ws + WS_H_OFF;           // (64, 256)

    // ---- Preload time-invariant weight columns into VGPRs (B operands) ----
    v2f bwx[64], bwh[64];
#pragma unroll
    for (int kk = 0; kk < 64; ++kk) {
        const int k = kk * 4 + koff;
        v2f a, b;
        a.x = Wx[(size_t)(k    ) * G4 + n0 + l16];
        a.y = Wx[(size_t)(k + 1) * G4 + n0 + l16];
        b.x = Wh[(size_t)(k    ) * G4 + n0 + l16];
        b.y = Wh[(size_t)(k + 1) * G4 + n0 + l16];
        bwx[kk] = a;
        bwh[kk] = b;
    }
    const float bias = Bx[n0 + l16] + Bh[n0 + l16];

    const v8i g1x = tdm_g1(SEQ_T * FEAT);    // x row stride (elements)
    const v8i g1h = tdm_g1(HID);             // h row stride

    const float* sh_xrow = sh_x + l16 * LDS_ROW + koff;
    const float* sh_hrow = sh_h + l16 * LDS_ROW + koff;

    // cell state: 2 elements per thread, in registers for the whole scan
    float c0 = 0.0f, c1 = 0.0f;

    // ---- Preamble: stage x(0), compute accX(0) = bias + x_0 @ Wx ----
#if USE_TDM
    if (w == 0) {
        tdm_issue(tdm_g0(LDSX_OFF, x + ((size_t)m0 * SEQ_T + 0) * FEAT), g1x);
        __builtin_amdgcn_s_wait_tensorcnt(0);
    }
#else
    coop_copy_tile(sh_x, x + ((size_t)m0 * SEQ_T + 0) * FEAT, (size_t)SEQ_T * FEAT);
#endif
    __syncthreads();

    v8f accX;
    {
        v8f a0, a1;
#pragma unroll
        for (int r = 0; r < 8; ++r) { a0[r] = bias; a1[r] = 0.0f; }
#pragma unroll
        for (int kk = 0; kk < 64; kk += 2) {    // two independent WMMA chains
            v2f f0 = *(const v2f*)(sh_xrow + kk * 4);
            v2f f1 = *(const v2f*)(sh_xrow + kk * 4 + 4);
            a0 = __builtin_amdgcn_wmma_f32_16x16x4_f32(
                false, f0, false, bwx[kk],     (short)0, a0, false, false);
            a1 = __builtin_amdgcn_wmma_f32_16x16x4_f32(
                false, f1, false, bwx[kk + 1], (short)0, a1, false, false);
        }
#pragma unroll
        for (int r = 0; r < 8; ++r) accX[r] = a0[r] + a1[r];
    }

    for (int t = 0; t < SEQ_T; ++t) {
        // ---- Stage h(t) (and prefetch x(t+1)) into LDS ----
        __syncthreads();                     // sh_x / sh_h free for rewrite
#if USE_TDM
        if (w == 0) {
            if (t + 1 < SEQ_T)
                tdm_issue(tdm_g0(LDSX_OFF, x + ((size_t)m0 * SEQ_T + (t + 1)) * FEAT), g1x);
            tdm_issue(tdm_g0(LDSH_OFF, hbuf + (size_t)m0 * HID), g1h);
            __builtin_amdgcn_s_wait_tensorcnt(0);
        }
#else
        if (t + 1 < SEQ_T)
            coop_copy_tile(sh_x, x + ((size_t)m0 * SEQ_T + (t + 1)) * FEAT, (size_t)SEQ_T * FEAT);
        coop_copy_tile(sh_h, hbuf + (size_t)m0 * HID, HID);
#endif
        __syncthreads();

        // ---- Phase 1 (critical path): gates tile = accX + h @ Wh ----
        v8f acc;
        {
            v8f a0 = accX, a1;
#pragma unroll
            for (int r = 0; r < 8; ++r) a1[r] = 0.0f;
#pragma unroll
            for (int kk = 0; kk < 64; kk += 2) {  // two independent WMMA chains
                v2f f0 = *(const v2f*)(sh_hrow + kk * 4);
                v2f f1 = *(const v2f*)(sh_hrow + kk * 4 + 4);
                a0 = __builtin_amdgcn_wmma_f32_16x16x4_f32(
                    false, f0, false, bwh[kk],     (short)0, a0, false, false);
                a1 = __builtin_amdgcn_wmma_f32_16x16x4_f32(
                    false, f1, false, bwh[kk + 1], (short)0, a1, false, false);
            }
#pragma unroll
            for (int r = 0; r < 8; ++r) acc[r] = a0[r] + a1[r];
        }

        // D layout: VGPR r -> row r (lanes 0-15) / r+8 (lanes 16-31), col = l16
#pragma unroll
        for (int r = 0; r < 8; ++r)
            sh_g[w * GPITCH + (r + half * 8) * 17 + l16] = acc[r];
        __syncthreads();

        // ---- Phase 2 (block-local): gate fusion, c in registers, publish h ----
        {
            const int e0 = threadIdx.x * 2;  // 2 of 256 tile elements
#pragma unroll
            for (int e = 0; e < 2; ++e) {
                const int idx = e0 + e;
                const int row = idx >> 4;
                const int col = idx & 15;
                const int o   = row * 17 + col;
                const float ig = fast_sigmoid(sh_g[              o]);
                const float fg = fast_sigmoid(sh_g[    GPITCH + o]);
                const float gg = fast_tanh  (sh_g[2 * GPITCH + o]);
                const float og = fast_sigmoid(sh_g[3 * GPITCH + o]);
                const float cold = e ? c1 : c0;
                const float cnew = fg * cold + ig * gg;
                const float hnew = og * fast_tanh(cnew);
                if (e) c1 = cnew; else c0 = cnew;
                const int brow = m0 + row;
                const int hid  = ht * 16 + col;
                hbuf[(size_t)brow * HID + hid] = hnew;
                out[((size_t)brow * SEQ_T + t) * HID + hid] = hnew;
            }
        }

        // ---- Off h-dependency chain: accX(t+1) = bias + x_{t+1} @ Wx ----
        if (t + 1 < SEQ_T) {
            v8f a0, a1;
#pragma unroll
            for (int r = 0; r < 8; ++r) { a0[r] = bias; a1[r] = 0.0f; }
#pragma unroll
            for (int kk = 0; kk < 64; kk += 2) {
                v2f f0 = *(const v2f*)(sh_xrow + kk * 4);
                v2f f1 = *(const v2f*)(sh_xrow + kk * 4 + 4);
                a0 = __builtin_amdgcn_wmma_f32_16x16x4_f32(
                    false, f0, false, bwx[kk],     (short)0, a0, false, false);
                a1 = __builtin_amdgcn_wmma_f32_16x16x4_f32(
                    false, f1, false, bwx[kk + 1], (short)0, a1, false, false);
            }
#pragma unroll
            for (int r = 0; r < 8; ++r) accX[r] = a0[r] + a1[r];
        }

        // ---- One 16-block barrier per step: h published to mt-group ----
        __threadfence();
        group_barrier(ctr, (unsigned)(t + 1) * GROUP_BLOCKS);
    }

    // ---- h_n tail: each (b, hid) element owned by exactly one block ----
#pragma unroll
    for (int e = 0; e < 2; ++e) {
        const int idx  = threadIdx.x * 2 + e;
        const int brow = m0 + (idx >> 4);
        const int hid  = ht * 16 + (idx & 15);
        out[(size_t)BATCH * SEQ_T * HID + (size_t)brow * HID + hid] =
            hbuf[(size_t)brow * HID + hid];
    }
}

extern "C" void kernel_launch(void* const* d_in, const int* in_sizes, int n_in,
                              void* d_out, int out_size, void* d_ws, size_t ws_size,
                              hipStream_t stream) {
    const float* x  = (const float*)d_in[0];
    const float* Wx = (const float*)d_in[1];
    const float* Wh = (const float*)d_in[2];
    const float* Bx = (const float*)d_in[3];
    const float* Bh = (const float*)d_in[4];
    float* out = (float*)d_out;
    float* ws  = (float*)d_ws;

    // Deterministic per call & graph-capturable: zero barrier counters + h0.
    hipMemsetAsync(d_ws, 0, WS_ZERO_BYTES, stream);

    lstm_persistent<<<NBLOCKS, TPB, LDS_BYTES, stream>>>(x, Wx, Wh, Bx, Bh, out, ws);
}